// SpatialGrid3D_21234318312198
// MI455X (gfx1250) — compile-verified
//
#include <hip/hip_runtime.h>
#include <hip/hip_bf16.h>
#include <stdint.h>

// ---------------------------------------------------------------------------
// Trilinear grid interpolation, 4M points into a 128^3 x 8 fp32 table.
// Memory-bound gather kernel: keep the 64MB table resident in the 192MB L2,
// stream coords/output with non-temporal hints, gather with b128 loads.
// Coordinates are staged into LDS via CDNA5 async global->LDS copies
// (GLOBAL_LOAD_ASYNC_TO_LDS_B128, ASYNCcnt) when the builtin is available.
// ---------------------------------------------------------------------------

#ifndef __has_builtin
#define __has_builtin(x) 0
#endif

#define BLOCK 256
#define PPT 2                   // points per thread
#define PPB (BLOCK * PPT)       // 512 points per block
#define GRIDN 128
#define LAT 8                   // latent channels (floats per cell)
#define SY (GRIDN * LAT)        // +1 in y, in floats
#define SZ (GRIDN * GRIDN * LAT) // +1 in z, in floats

typedef float v4f __attribute__((ext_vector_type(4)));
typedef int   v4i __attribute__((ext_vector_type(4)));

#if __has_builtin(__builtin_amdgcn_global_load_async_to_lds_b128)
#define HAVE_ASYNC_LDS 1
// global (addrspace(1)) and LDS (addrspace(3)) int4 pointers for the builtin
typedef __attribute__((address_space(1))) v4i gv4i;
typedef __attribute__((address_space(3))) v4i lv4i;
#else
#define HAVE_ASYNC_LDS 0
#endif

__device__ __forceinline__ v4f ld4(const float* p) {
  return *reinterpret_cast<const v4f*>(p);
}

// matches reference: p*(1-t) + t*q
__device__ __forceinline__ v4f lerp4(v4f p, v4f q, float t, float omt) {
  return p * omt + q * t;
}

__device__ __forceinline__ void process_point(float u, float v, float w,
                                              const float* __restrict__ table,
                                              float* __restrict__ outp) {
  float fx = u * 127.0f, fy = v * 127.0f, fz = w * 127.0f;
  int ix = (int)fx, iy = (int)fy, iz = (int)fz;   // coords in [0,1) -> trunc ok
  float a = fx - (float)ix;
  float b = fy - (float)iy;
  float g = fz - (float)iz;
  float oma = 1.0f - a, omb = 1.0f - b, omg = 1.0f - g;

  const float* base = table + (size_t)(((iz * GRIDN) + iy) * GRIDN + ix) * LAT;

  v4f rlo[2], rhi[2];
#pragma unroll
  for (int z = 0; z < 2; ++z) {
    const float* bz = base + z * SZ;
    // x-lerp for y=0 and y=1 rows (each cell = 32B = 2 x b128, RT policy so the
    // table stays hot in L2)
    v4f y0lo = lerp4(ld4(bz),          ld4(bz + LAT),          a, oma);
    v4f y0hi = lerp4(ld4(bz + 4),      ld4(bz + LAT + 4),      a, oma);
    v4f y1lo = lerp4(ld4(bz + SY),     ld4(bz + SY + LAT),     a, oma);
    v4f y1hi = lerp4(ld4(bz + SY + 4), ld4(bz + SY + LAT + 4), a, oma);
    rlo[z] = lerp4(y0lo, y1lo, b, omb);
    rhi[z] = lerp4(y0hi, y1hi, b, omb);
  }
  v4f olo = lerp4(rlo[0], rlo[1], g, omg);
  v4f ohi = lerp4(rhi[0], rhi[1], g, omg);

  // streaming output: non-temporal so 128MB of writes don't evict the table
  __builtin_nontemporal_store(olo, (v4f*)outp);
  __builtin_nontemporal_store(ohi, (v4f*)(outp + 4));
}

__global__ __launch_bounds__(BLOCK) void trilerp_kernel(
    const float* __restrict__ uv, const float* __restrict__ table,
    float* __restrict__ out, int n) {
  __shared__ __align__(16) float spts[PPB * 3];  // 6144 B of staged coords

  const int t = threadIdx.x;
  const int blockStart = blockIdx.x * PPB;             // first point of block
  const int rem = n - blockStart;
  const int npts = rem > PPB ? PPB : rem;              // points this block
  const int nbytes = npts * 12;                        // coord bytes to stage
  const int chunks = nbytes >> 4;                      // 16B chunks

  // ---- stage coordinates into LDS (coalesced 16B/lane) ----
#if HAVE_ASYNC_LDS
  {
    gv4i* g = (gv4i*)(uintptr_t)((const char*)uv + (size_t)blockStart * 12);
    lv4i* l = (lv4i*)(unsigned)(uintptr_t)(&spts[0]);
    for (int c = t; c < chunks; c += BLOCK) {
      __builtin_amdgcn_global_load_async_to_lds_b128(g + c, l + c, 0, 0);
    }
#if __has_builtin(__builtin_amdgcn_s_wait_asynccnt)
    __builtin_amdgcn_s_wait_asynccnt(0);
#else
    asm volatile("s_wait_asynccnt 0" ::: "memory");
#endif
  }
#else
  {
    const v4f* gsrc = (const v4f*)((const char*)uv + (size_t)blockStart * 12);
    for (int c = t; c < chunks; c += BLOCK) {
      ((v4f*)spts)[c] = __builtin_nontemporal_load(gsrc + c);
    }
  }
#endif
  // tail floats (only when npts % 4 != 0; never hit for this problem size)
  for (int f = chunks * 4 + t; f < (nbytes >> 2); f += BLOCK) {
    spts[f] = uv[(size_t)blockStart * 3 + f];
  }
  __syncthreads();

  // ---- gather + blend, 2 points per thread, stores stay wave-coalesced ----
#pragma unroll
  for (int k = 0; k < PPT; ++k) {
    int p = t + k * BLOCK;
    int i = blockStart + p;
    if (i < n) {
      float u = spts[3 * p + 0];
      float v = spts[3 * p + 1];
      float w = spts[3 * p + 2];
      process_point(u, v, w, table, out + (size_t)i * LAT);
    }
  }
}

extern "C" void kernel_launch(void* const* d_in, const int* in_sizes, int n_in,
                              void* d_out, int out_size, void* d_ws,
                              size_t ws_size, hipStream_t stream) {
  const float* uv = (const float*)d_in[0];     // N_PTS x 3 fp32
  const float* table = (const float*)d_in[1];  // 128^3 x 8 fp32
  float* out = (float*)d_out;                  // N_PTS x 8 fp32
  int n = in_sizes[0] / 3;
  int blocks = (n + PPB - 1) / PPB;
  trilerp_kernel<<<blocks, BLOCK, 0, stream>>>(uv, table, out, n);
}